// OneToOneLinearStackedConv2d_24945170055342
// MI455X (gfx1250) — compile-verified
//
#include <hip/hip_runtime.h>

typedef __attribute__((ext_vector_type(2))) float v2f;
typedef __attribute__((ext_vector_type(8))) float v8f;

#define CIN   8
#define COUT  16
#define HW    224
#define TH    16
#define TW    16
// patch 24x24 -> conv1 22x22 -> dw1 20x20 -> dw2 18x18 -> final 16x16

// LDS layout (float offsets); buffers aliased by liveness:
#define OFF_IN   0        // [8][24][24] = 4608   (dead after phase 1)
#define OFF_W0T  4608     // [72][16]    = 1152   (dead after phase 1)
#define OFF_C2   0        // [16][20][20]= 6400   (phase 2+, aliases IN/W0T)
#define OFF_C1   6400     // [16][22*22] = 7744   (dead after phase 2)
#define OFF_C3   6400     // [16][18][18]= 5184   (phase 3+, aliases C1)
#define SMEM_F   14144    // 56,576 bytes

__global__ __launch_bounds__(256)
void fused_stackedconv_kernel(const float* __restrict__ x,
                              const float* __restrict__ W0,
                              const float* __restrict__ W1,
                              const float* __restrict__ W2,
                              const float* __restrict__ W3,
                              const float* __restrict__ b3,
                              float* __restrict__ out)
{
    __shared__ float smem[SMEM_F];
    const int tid  = threadIdx.x;
    const int lane = tid & 31;
    const int wid  = tid >> 5;
    const int ox = blockIdx.x * TW;
    const int oy = blockIdx.y * TH;
    const int bb = blockIdx.z;

    // ---- phase 0a: W0 transposed to K-major [k=72][n=16] in LDS
    for (int idx = tid; idx < 72 * 16; idx += 256) {
        const int k = idx >> 4, n = idx & 15;
        smem[OFF_W0T + idx] = W0[n * 72 + k];   // W0 is [Cout][Cin*9]
    }
    // ---- phase 0b: zero-padded input patch [8][24][24]
    const float* xb = x + (size_t)bb * CIN * HW * HW;
    for (int idx = tid; idx < CIN * 24 * 24; idx += 256) {
        const int c  = idx / 576, rem = idx % 576;
        const int yy = rem / 24,  xx  = rem % 24;
        const int gy = oy - 4 + yy, gx = ox - 4 + xx;
        float v = 0.0f;
        if (gy >= 0 && gy < HW && gx >= 0 && gx < HW)
            v = xb[(c * HW + gy) * HW + gx];
        smem[OFF_IN + idx] = v;
    }
    __syncthreads();

    // ---- phase 1: dense conv (Cin*9 = 72 = 18 x K4) via V_WMMA_F32_16X16X4_F32
    // 484 conv1 outputs linearized (p = i*22 + j); 31 M-tiles of 16 positions
    // (vs 44 with per-row chunking: 30% fewer WMMAs + A loads, contiguous C stores).
    const int r  = lane >> 4;    // half-wave: selects K pair per ISA A/B layouts
    const int nl = lane & 15;    // N (out channel) for B/C; M row for A

    // Per-lane B fragments + A LDS-offset tables, hoisted out of the tile loop.
    v2f bfr[18];
    int offx[18], offy[18];
#pragma unroll
    for (int kk = 0; kk < 18; ++kk) {
        const int kx = 4 * kk + 2 * r;   // A/B VGPR0 K-index for this half-wave
        const int ky = kx + 1;           // A/B VGPR1 K-index
        bfr[kk].x = smem[OFF_W0T + kx * 16 + nl];
        bfr[kk].y = smem[OFF_W0T + ky * 16 + nl];
        const int cinx = kx / 9, tapx = kx - cinx * 9;
        const int ciny = ky / 9, tapy = ky - ciny * 9;
        offx[kk] = cinx * 576 + (tapx / 3) * 24 + (tapx % 3);
        offy[kk] = ciny * 576 + (tapy / 3) * 24 + (tapy % 3);
    }

    for (int t = wid; t < 31; t += 8) {
        const int p  = t * 16 + nl;           // this lane's A-row position
        const int pp = p < 483 ? p : 483;     // clamp reads in the ragged last tile
        const int i  = pp / 22;
        const int j  = pp - i * 22;
        const int abase = OFF_IN + i * 24 + j;

        v8f acc = {};
#pragma unroll
        for (int kk = 0; kk < 18; ++kk) {
            v2f a;
            a.x = smem[abase + offx[kk]];
            a.y = smem[abase + offy[kk]];
            acc = __builtin_amdgcn_wmma_f32_16x16x4_f32(
                false, a, false, bfr[kk], (short)0, acc, false, false);
        }
        // C layout: VGPR v -> M = v + 8r; N = nl.  Global position = t*16 + M,
        // so the 8 results per lane are CONSECUTIVE in LDS -> wide ds stores.
        const int pbase = t * 16 + 8 * r;
        const int cdst  = OFF_C1 + nl * 484 + pbase;
        if (t < 30) {
#pragma unroll
            for (int v = 0; v < 8; ++v)
                smem[cdst + v] = acc[v];
        } else {
#pragma unroll
            for (int v = 0; v < 8; ++v)
                if (pbase + v < 484)
                    smem[cdst + v] = acc[v];
        }
    }
    __syncthreads();

    // ---- phase 2: depthwise 1  [16][22][22] -> [16][20][20]
    for (int idx = tid; idx < 16 * 20 * 20; idx += 256) {
        const int ch = idx / 400, rem = idx % 400;
        const int yy = rem / 20,  xx  = rem % 20;
        const float* w = W1 + ch * 9;
        float s = 0.0f;
#pragma unroll
        for (int ty = 0; ty < 3; ++ty)
#pragma unroll
            for (int tx = 0; tx < 3; ++tx)
                s += smem[OFF_C1 + ch * 484 + (yy + ty) * 22 + (xx + tx)] * w[ty * 3 + tx];
        smem[OFF_C2 + idx] = s;
    }
    __syncthreads();

    // ---- phase 3: depthwise 2  [16][20][20] -> [16][18][18]
    for (int idx = tid; idx < 16 * 18 * 18; idx += 256) {
        const int ch = idx / 324, rem = idx % 324;
        const int yy = rem / 18,  xx  = rem % 18;
        const float* w = W2 + ch * 9;
        float s = 0.0f;
#pragma unroll
        for (int ty = 0; ty < 3; ++ty)
#pragma unroll
            for (int tx = 0; tx < 3; ++tx)
                s += smem[OFF_C2 + ch * 400 + (yy + ty) * 20 + (xx + tx)] * w[ty * 3 + tx];
        smem[OFF_C3 + idx] = s;
    }
    __syncthreads();

    // ---- phase 4: depthwise 3 + bias -> global [16][16][16]
    for (int idx = tid; idx < 16 * 16 * 16; idx += 256) {
        const int ch = idx >> 8, rem = idx & 255;
        const int yy = rem >> 4, xx = rem & 15;
        const float* w = W3 + ch * 9;
        float s = b3[ch];
#pragma unroll
        for (int ty = 0; ty < 3; ++ty)
#pragma unroll
            for (int tx = 0; tx < 3; ++tx)
                s += smem[OFF_C3 + ch * 324 + (yy + ty) * 18 + (xx + tx)] * w[ty * 3 + tx];
        out[((size_t)(bb * COUT + ch) * HW + (oy + yy)) * HW + (ox + xx)] = s;
    }
}

extern "C" void kernel_launch(void* const* d_in, const int* in_sizes, int n_in,
                              void* d_out, int out_size, void* d_ws, size_t ws_size,
                              hipStream_t stream) {
    const float* x  = (const float*)d_in[0];
    const float* W0 = (const float*)d_in[1];
    const float* W1 = (const float*)d_in[2];
    const float* W2 = (const float*)d_in[3];
    const float* W3 = (const float*)d_in[4];
    const float* b3 = (const float*)d_in[5];
    float* out = (float*)d_out;

    dim3 grid(HW / TW, HW / TH, 32);   // 14 x 14 x 32 blocks
    dim3 block(256);                   // 8 waves (wave32)
    fused_stackedconv_kernel<<<grid, block, 0, stream>>>(x, W0, W1, W2, W3, b3, out);
}